// LateralInhibitionAttention_53051436040624
// MI455X (gfx1250) — compile-verified
//
#include <hip/hip_runtime.h>

typedef __attribute__((ext_vector_type(2))) float v2f;
typedef __attribute__((ext_vector_type(8))) float v8f;
typedef __attribute__((address_space(3))) float lds_float;

#define Bn 4
#define Nn 1024
#define Cn 768
#define Hn 12
#define Dn 64
#define QKV_ONE (Bn*Hn*Nn*Dn)   // 3145728 elements per q/k/v tensor

// 32-bit LDS byte offset of a generic pointer known to live in LDS
__device__ __forceinline__ unsigned lds_off(const float* p) {
  return (unsigned)(size_t)(lds_float*)p;
}

// async 16B/lane global->LDS copy (gfx1250 GLOBAL_LOAD_ASYNC_TO_LDS_B128, ASYNCcnt)
__device__ __forceinline__ void async_copy_b128(const float* gsrc, const float* ldst) {
  unsigned l = lds_off(ldst);
  asm volatile("global_load_async_to_lds_b128 %0, %1, off"
               :: "v"(l), "v"(gsrc) : "memory");
}
__device__ __forceinline__ void async_wait0() {
  asm volatile("s_wait_asynccnt 0x0" ::: "memory");
}

// ---------------------------------------------------------------------------
// Kernel 1: LayerNorm over C=768, one row per block (256 threads, 3 elems/thr)
// ---------------------------------------------------------------------------
__global__ __launch_bounds__(256) void ln_kernel(const float* __restrict__ x,
                                                 const float* __restrict__ g,
                                                 const float* __restrict__ bta,
                                                 float* __restrict__ xn) {
  __shared__ float red[256];
  const int row = blockIdx.x;
  const int t = threadIdx.x;
  const float* xr = &x[row * Cn];
  float v0 = xr[t], v1 = xr[t + 256], v2 = xr[t + 512];
  red[t] = v0 + v1 + v2;
  __syncthreads();
  for (int off = 128; off; off >>= 1) {
    if (t < off) red[t] += red[t + off];
    __syncthreads();
  }
  float mu = red[0] * (1.0f / 768.0f);
  __syncthreads();
  float d0 = v0 - mu, d1 = v1 - mu, d2 = v2 - mu;
  red[t] = d0 * d0 + d1 * d1 + d2 * d2;
  __syncthreads();
  for (int off = 128; off; off >>= 1) {
    if (t < off) red[t] += red[t + off];
    __syncthreads();
  }
  float rs = rsqrtf(red[0] * (1.0f / 768.0f) + 1e-5f);
  float* o = &xn[row * Cn];
  o[t]       = d0 * rs * g[t]       + bta[t];
  o[t + 256] = d1 * rs * g[t + 256] + bta[t + 256];
  o[t + 512] = d2 * rs * g[t + 512] + bta[t + 512];
}

// ---------------------------------------------------------------------------
// Kernel 2: QKV projection GEMM  out[r,j] = xn[r,:] . W[j,:] + bias[j]
// NT GEMM, fp32 WMMA 16x16x4, 64x64 tile per 256-thread (8 wave) block.
// A/W tiles staged via async global->LDS (no VGPR round trip).
// ---------------------------------------------------------------------------
#define K2_LDA 36   // 32 + 4 pad; 144B row stride (16B aligned)
__global__ __launch_bounds__(256) void qkv_gemm(const float* __restrict__ xn,
                                                const float* __restrict__ W,
                                                const float* __restrict__ bias,
                                                float* __restrict__ qkv) {
  __shared__ float As[64 * K2_LDA];
  __shared__ float Ws[64 * K2_LDA];
  const int t = threadIdx.x;
  const int lane = t & 31;
  const int wave = t >> 5;
  const int l16 = lane & 15;
  const int hi = lane >> 4;           // lane halves hold K pairs {0,1}/{2,3}
  const int rbase = blockIdx.x * 64;
  const int jbase = blockIdx.y * 64;
  const int mt = wave & 3;            // M subtile 0..3
  const int ntBase = (wave >> 2) * 2; // N subtiles {0,1} or {2,3}

  v8f acc0, acc1;
  {
    float b0 = bias[jbase + ntBase * 16 + l16];
    float b1 = bias[jbase + (ntBase + 1) * 16 + l16];
    for (int i = 0; i < 8; ++i) { acc0[i] = b0; acc1[i] = b1; }
  }

  for (int kb = 0; kb < Cn; kb += 32) {
    for (int i = t; i < 64 * 8; i += 256) {          // 512 b128 copies each buf
      int r = i >> 3, c4 = i & 7;
      async_copy_b128(&xn[(rbase + r) * Cn + kb + c4 * 4], &As[r * K2_LDA + c4 * 4]);
      async_copy_b128(&W[(jbase + r) * Cn + kb + c4 * 4],  &Ws[r * K2_LDA + c4 * 4]);
    }
    async_wait0();
    __syncthreads();
    for (int kk = 0; kk < 32; kk += 4) {
      v2f a  = *(const v2f*)&As[(mt * 16 + l16) * K2_LDA + kk + hi * 2];
      v2f b0 = *(const v2f*)&Ws[(ntBase * 16 + l16) * K2_LDA + kk + hi * 2];
      v2f b1 = *(const v2f*)&Ws[((ntBase + 1) * 16 + l16) * K2_LDA + kk + hi * 2];
      acc0 = __builtin_amdgcn_wmma_f32_16x16x4_f32(false, a, false, b0, (short)0, acc0, false, false);
      acc1 = __builtin_amdgcn_wmma_f32_16x16x4_f32(false, a, false, b1, (short)0, acc1, false, false);
    }
    __syncthreads();
  }

  for (int pair = 0; pair < 2; ++pair) {
    v8f acc = pair ? acc1 : acc0;
    int j = jbase + (ntBase + pair) * 16 + l16;
    int which = j / 768, rem = j % 768;
    int h = rem >> 6, d = rem & 63;
    for (int vv = 0; vv < 8; ++vv) {
      int r = rbase + mt * 16 + vv + hi * 8;
      int b = r >> 10, n = r & 1023;
      qkv[which * QKV_ONE + (((b * Hn + h) << 10) + n) * Dn + d] = acc[vv];
    }
  }
}

// ---------------------------------------------------------------------------
// Kernel 3: fused attention per (b,h, 16-row query tile).  Block = 128 thr.
//   - Q band (16 rows + 2 halo rows) staged in LDS
//   - S band 18x1024 built in LDS: WMMA for the 16 main rows, VALU dots for
//     the 2 halo rows (zero at grid edges == 'SAME' zero padding)
//   - 3x3 ones-minus-center conv, attn' = S - 0.2*neigh  -> P in LDS
//   - wave32 shuffle softmax per row
//   - P @ V with WMMA, direct write to [B,N,C] output
//   K/V chunks staged via async global->LDS.
// ---------------------------------------------------------------------------
#define QS_LD 68
#define KS_LD 68
#define S_LD  1026
__global__ __launch_bounds__(128) void attn_fused(const float* __restrict__ q,
                                                  const float* __restrict__ kmat,
                                                  const float* __restrict__ vmat,
                                                  float* __restrict__ out) {
  extern __shared__ float sm[];
  float* Qs = sm;                    // 18*68
  float* Ks = Qs + 18 * QS_LD;       // 64*68 (reused for V chunks)
  float* S  = Ks + 64 * KS_LD;       // 18*1026
  float* P  = S + 18 * S_LD;         // 16*1024

  const int t = threadIdx.x, lane = t & 31, wave = t >> 5;
  const int l16 = lane & 15, hi = lane >> 4;
  const int bh = blockIdx.x >> 6;    // 0..47
  const int qt = blockIdx.x & 63;
  const int m0 = qt << 4;
  const int b = bh / Hn, h = bh % Hn;
  const float scale = 0.125f;        // D^-0.5

  // stage Q band (rows m0-1 .. m0+16, zero-filled out of range)
  for (int i = t; i < 18 * 64; i += 128) {
    int r = i >> 6, d = i & 63;
    int gr = m0 - 1 + r;
    Qs[r * QS_LD + d] = (gr >= 0 && gr < Nn) ? q[(bh * Nn + gr) * Dn + d] : 0.0f;
  }
  __syncthreads();

  // ---- scores: S[band 0..17][0..1023] ----
  for (int nb = 0; nb < 16; ++nb) {
    for (int i = t; i < 64 * 16; i += 128) {           // stage 64 K rows
      int r = i >> 4, c4 = i & 15;
      async_copy_b128(&kmat[(bh * Nn + nb * 64 + r) * Dn + c4 * 4],
                      &Ks[r * KS_LD + c4 * 4]);
    }
    async_wait0();
    __syncthreads();

    // main 16 rows via WMMA: wave w -> cols nb*64 + w*16 ..
    v8f acc; for (int i = 0; i < 8; ++i) acc[i] = 0.0f;
    for (int kk = 0; kk < 64; kk += 4) {
      v2f a  = *(const v2f*)&Qs[(1 + l16) * QS_LD + kk + hi * 2];
      v2f bf = *(const v2f*)&Ks[(wave * 16 + l16) * KS_LD + kk + hi * 2];
      acc = __builtin_amdgcn_wmma_f32_16x16x4_f32(false, a, false, bf, (short)0, acc, false, false);
    }
    {
      int col = nb * 64 + wave * 16 + l16;
      for (int vv = 0; vv < 8; ++vv)
        S[(1 + vv + hi * 8) * S_LD + col] = acc[vv] * scale;
    }
    // 2 halo rows via VALU dots: 128 threads == 2 rows x 64 cols
    {
      int hr = t >> 6, n = t & 63;
      int bandr = hr ? 17 : 0;
      float s = 0.0f;
      for (int kk = 0; kk < 64; ++kk)
        s += Qs[bandr * QS_LD + kk] * Ks[n * KS_LD + kk];
      S[bandr * S_LD + nb * 64 + n] = s * scale;
    }
    __syncthreads();
  }

  // ---- 3x3 lateral inhibition conv, P = S - 0.2*neigh ----
  for (int idx = t; idx < 16 * 1024; idx += 128) {
    int r = idx >> 10, n = idx & 1023;
    const float* S0 = &S[r * S_LD + n];
    const float* S1 = S0 + S_LD;
    const float* S2 = S1 + S_LD;
    float l0 = n ? S0[-1] : 0.f, r0 = (n < 1023) ? S0[1] : 0.f;
    float l1 = n ? S1[-1] : 0.f, r1 = (n < 1023) ? S1[1] : 0.f;
    float l2 = n ? S2[-1] : 0.f, r2 = (n < 1023) ? S2[1] : 0.f;
    float neigh = l0 + S0[0] + r0 + l1 + r1 + l2 + S2[0] + r2;
    P[r * 1024 + n] = S1[0] - 0.2f * neigh;
  }
  __syncthreads();

  // ---- softmax: each wave owns rows {wave, wave+4, wave+8, wave+12} ----
  for (int rr = 0; rr < 4; ++rr) {
    int row = wave + rr * 4;
    float* pr = &P[row * 1024];
    float m = -3.402823466e38f;
    for (int i = lane; i < 1024; i += 32) m = fmaxf(m, pr[i]);
    for (int off = 16; off; off >>= 1) m = fmaxf(m, __shfl_xor(m, off, 32));
    float s = 0.0f;
    for (int i = lane; i < 1024; i += 32) {
      float e = __expf(pr[i] - m);
      pr[i] = e;
      s += e;
    }
    for (int off = 16; off; off >>= 1) s += __shfl_xor(s, off, 32);
    float inv = 1.0f / s;
    for (int i = lane; i < 1024; i += 32) pr[i] *= inv;
  }
  __syncthreads();

  // ---- P @ V with WMMA: wave w -> output cols w*16..w*16+15 ----
  v8f oacc; for (int i = 0; i < 8; ++i) oacc[i] = 0.0f;
  for (int kc = 0; kc < 16; ++kc) {
    for (int i = t; i < 64 * 16; i += 128) {           // stage 64 V rows
      int r = i >> 4, c4 = i & 15;
      async_copy_b128(&vmat[(bh * Nn + kc * 64 + r) * Dn + c4 * 4],
                      &Ks[r * KS_LD + c4 * 4]);
    }
    async_wait0();
    __syncthreads();
    for (int kk = 0; kk < 64; kk += 4) {
      int kcol = kc * 64 + kk;
      v2f a = *(const v2f*)&P[l16 * 1024 + kcol + hi * 2];
      v2f bf;
      int kl = kk + hi * 2;
      bf.x = Ks[(kl + 0) * KS_LD + wave * 16 + l16];
      bf.y = Ks[(kl + 1) * KS_LD + wave * 16 + l16];
      oacc = __builtin_amdgcn_wmma_f32_16x16x4_f32(false, a, false, bf, (short)0, oacc, false, false);
    }
    __syncthreads();
  }
  {
    int col = h * 64 + wave * 16 + l16;
    for (int vv = 0; vv < 8; ++vv) {
      int n = m0 + vv + hi * 8;
      out[(b * Nn + n) * Cn + col] = oacc[vv];
    }
  }
}

// ---------------------------------------------------------------------------
extern "C" void kernel_launch(void* const* d_in, const int* in_sizes, int n_in,
                              void* d_out, int out_size, void* d_ws, size_t ws_size,
                              hipStream_t stream) {
  const float* x     = (const float*)d_in[0];
  const float* qkv_w = (const float*)d_in[1];
  const float* qkv_b = (const float*)d_in[2];
  const float* ln_g  = (const float*)d_in[3];
  const float* ln_b  = (const float*)d_in[4];
  float* out = (float*)d_out;
  float* ws  = (float*)d_ws;

  float* xn  = ws;                         // 4096*768
  float* qkv = ws + (size_t)Bn * Nn * Cn;  // 3*QKV_ONE

  ln_kernel<<<Bn * Nn, 256, 0, stream>>>(x, ln_g, ln_b, xn);
  qkv_gemm<<<dim3(64, 36), 256, 0, stream>>>(xn, qkv_w, qkv_b, qkv);

  const float* q = qkv;
  const float* k = qkv + QKV_ONE;
  const float* v = qkv + 2 * QKV_ONE;
  size_t smem = (size_t)(18 * QS_LD + 64 * KS_LD + 18 * S_LD + 16 * 1024) * sizeof(float);
  attn_fused<<<Bn * Hn * (Nn / 16), 128, smem, stream>>>(q, k, v, out);
}